// Point_Volumetric_Renderer_55490977464805
// MI455X (gfx1250) — compile-verified
//
#include <hip/hip_runtime.h>

typedef __attribute__((ext_vector_type(2))) float v2f;
typedef __attribute__((ext_vector_type(8))) float v8f;

constexpr int kC   = 16;          // feature channels
constexpr int kK   = 8;           // neighbors
constexpr int kR   = 4096;        // rays
constexpr int kSR  = 128;         // samples per ray
constexpr int kN   = kR * kSR;    // 524288 sample points
constexpr int kPad = 17;          // LDS row stride (dwords), coprime with 64 banks

// ---------------------------------------------------------------------------
// Kernel 1: KNN inverse-distance feature aggregation + WMMA head
//   feat[n, 0:16] = sum_k w_nk * points_feat[idx_nk, :]
//   [rgb0 rgb1 rgb2 sigma] = feat @ W  (W = [w_rgb | w_sigma | 0...], 16x16)
// computed as 4 chained V_WMMA_F32_16X16X4_F32 per 16-point tile.
// ---------------------------------------------------------------------------
__global__ __launch_bounds__(256)
void pvr_feat_head(const float* __restrict__ points_feat,
                   const int*   __restrict__ indices,
                   const float* __restrict__ dists,
                   const float* __restrict__ w_rgb,
                   const float* __restrict__ w_sigma,
                   float* __restrict__ sigma_ws,   // [kN]
                   float* __restrict__ rgb_ws)     // [3][kN] planar
{
    __shared__ float sfeat[256 * kPad];

    const int tid = threadIdx.x;
    const int n   = blockIdx.x * 256 + tid;

    // ---- inverse-distance weights over K (vector loads: 32B aligned) ----
    const float4* dv = (const float4*)(dists + (size_t)n * kK);
    float4 da = dv[0], db = dv[1];
    float dd[kK] = {da.x, da.y, da.z, da.w, db.x, db.y, db.z, db.w};

    float w[kK];
    float wsum = 0.0f;
#pragma unroll
    for (int k = 0; k < kK; ++k) {
        w[k] = 1.0f / (dd[k] + 1e-7f);
        wsum += w[k];
    }
    const float inv = 1.0f / wsum;

    const int4* iv = (const int4*)(indices + (size_t)n * kK);
    int4 ia = iv[0], ib = iv[1];
    int id[kK] = {ia.x, ia.y, ia.z, ia.w, ib.x, ib.y, ib.z, ib.w};

    // ---- gather + weighted accumulate (L2-resident 32MB table) ----
    float acc[kC];
#pragma unroll
    for (int c = 0; c < kC; ++c) acc[c] = 0.0f;

#pragma unroll
    for (int k = 0; k < kK; ++k) {
        const float4* row = (const float4*)(points_feat + (size_t)id[k] * kC);
        const float wk = w[k] * inv;
        float4 f0 = row[0], f1 = row[1], f2 = row[2], f3 = row[3];
        acc[0]  = fmaf(wk, f0.x, acc[0]);  acc[1]  = fmaf(wk, f0.y, acc[1]);
        acc[2]  = fmaf(wk, f0.z, acc[2]);  acc[3]  = fmaf(wk, f0.w, acc[3]);
        acc[4]  = fmaf(wk, f1.x, acc[4]);  acc[5]  = fmaf(wk, f1.y, acc[5]);
        acc[6]  = fmaf(wk, f1.z, acc[6]);  acc[7]  = fmaf(wk, f1.w, acc[7]);
        acc[8]  = fmaf(wk, f2.x, acc[8]);  acc[9]  = fmaf(wk, f2.y, acc[9]);
        acc[10] = fmaf(wk, f2.z, acc[10]); acc[11] = fmaf(wk, f2.w, acc[11]);
        acc[12] = fmaf(wk, f3.x, acc[12]); acc[13] = fmaf(wk, f3.y, acc[13]);
        acc[14] = fmaf(wk, f3.z, acc[14]); acc[15] = fmaf(wk, f3.w, acc[15]);
    }

    // ---- stage feat tile to LDS (stride 17 -> conflict-free) ----
    float* dst = sfeat + tid * kPad;
#pragma unroll
    for (int c = 0; c < kC; ++c) dst[c] = acc[c];

    __syncthreads();

    // ---- WMMA head: D = A(16x16 feat, as 4 K-slices) x B(16x16 padded W) ----
    const int lane = tid & 31;
    const int wave = tid >> 5;
    const int half = lane >> 4;    // 0: lanes 0-15, 1: lanes 16-31
    const int nl   = lane & 15;    // column / row-in-half index

    // B operand: 4x16 f32 slice per step t (K rows kb..kb+3).
    // VGPR0 = K=kb+2*half, VGPR1 = K=kb+2*half+1, N = nl across lanes.
    v2f bmat[4];
#pragma unroll
    for (int t = 0; t < 4; ++t) {
        const int k0 = 4 * t + 2 * half;
        float b0 = 0.0f, b1 = 0.0f;
        if (nl < 3) {
            b0 = w_rgb[k0 * 3 + nl];
            b1 = w_rgb[(k0 + 1) * 3 + nl];
        } else if (nl == 3) {
            b0 = w_sigma[k0];
            b1 = w_sigma[k0 + 1];
        }
        bmat[t].x = b0;
        bmat[t].y = b1;
    }

#pragma unroll
    for (int tt = 0; tt < 2; ++tt) {
        const int rowBase = (wave * 2 + tt) * 16;   // tile start within block
        v8f d = {};                                  // C = 0 accumulator
#pragma unroll
        for (int t = 0; t < 4; ++t) {
            // A 16x4 f32 layout: lanes 0-15 -> K=kb,kb+1 ; lanes 16-31 -> K=kb+2,kb+3
            const int k0 = 4 * t + 2 * half;
            const float* src = sfeat + (rowBase + nl) * kPad + k0;
            v2f a;
            a.x = src[0];
            a.y = src[1];
            d = __builtin_amdgcn_wmma_f32_16x16x4_f32(
                    false, a, false, bmat[t], (short)0, d, false, false);
        }

        // D layout: VGPR v holds M = v + 8*half, N = nl. Only N<4 are real outputs.
        const int pbase = blockIdx.x * 256 + rowBase + 8 * half;
        if (nl < 3) {
#pragma unroll
            for (int v = 0; v < 8; ++v) {
                const float x = d[v];
                rgb_ws[(size_t)nl * kN + (pbase + v)] = 1.0f / (1.0f + __expf(-x));
            }
        } else if (nl == 3) {
#pragma unroll
            for (int v = 0; v < 8; ++v) {
                sigma_ws[pbase + v] = fmaxf(d[v], 0.0f);
            }
        }
    }
}

// ---------------------------------------------------------------------------
// Kernel 2: per-ray volumetric alpha compositing (serial cumprod over SR)
// ---------------------------------------------------------------------------
__global__ __launch_bounds__(128)
void pvr_composite(const float* __restrict__ sigma_ws,
                   const float* __restrict__ rgb_ws,
                   const float* __restrict__ delta,
                   const float* __restrict__ z_vals,
                   float* __restrict__ out)   // [kR][5]
{
    const int r = blockIdx.x * blockDim.x + threadIdx.x;
    if (r >= kR) return;

    float T = 1.0f;
    float r0 = 0.0f, r1 = 0.0f, r2 = 0.0f, depth = 0.0f, accw = 0.0f;
    const int base = r * kSR;

    for (int s = 0; s < kSR; ++s) {
        const int i = base + s;
        const float sig = sigma_ws[i];
        const float a   = 1.0f - __expf(-sig * delta[i]);
        const float wt  = a * T;
        r0    = fmaf(wt, rgb_ws[i],            r0);
        r1    = fmaf(wt, rgb_ws[kN + i],       r1);
        r2    = fmaf(wt, rgb_ws[2 * kN + i],   r2);
        depth = fmaf(wt, z_vals[i],            depth);
        accw += wt;
        T *= (1.0f - a + 1e-10f);
    }

    const float bg = 1.0f - accw;
    out[r * 5 + 0] = r0 + bg;
    out[r * 5 + 1] = r1 + bg;
    out[r * 5 + 2] = r2 + bg;
    out[r * 5 + 3] = depth;
    out[r * 5 + 4] = accw;
}

// ---------------------------------------------------------------------------
extern "C" void kernel_launch(void* const* d_in, const int* in_sizes, int n_in,
                              void* d_out, int out_size, void* d_ws, size_t ws_size,
                              hipStream_t stream)
{
    const float* points_feat = (const float*)d_in[0];
    const int*   indices     = (const int*)  d_in[1];
    const float* dists       = (const float*)d_in[2];
    const float* w_rgb       = (const float*)d_in[3];
    const float* w_sigma     = (const float*)d_in[4];
    const float* delta       = (const float*)d_in[5];
    const float* z_vals      = (const float*)d_in[6];
    float* out = (float*)d_out;

    float* sigma_ws = (float*)d_ws;        // kN floats
    float* rgb_ws   = sigma_ws + kN;       // 3*kN floats (planar)

    pvr_feat_head<<<kN / 256, 256, 0, stream>>>(
        points_feat, indices, dists, w_rgb, w_sigma, sigma_ws, rgb_ws);

    pvr_composite<<<kR / 128, 128, 0, stream>>>(
        sigma_ws, rgb_ws, delta, z_vals, out);
}